// GCN_EW_10943576670614
// MI455X (gfx1250) — compile-verified
//
#include <hip/hip_runtime.h>
#include <hip/hip_bf16.h>

typedef float v2f __attribute__((ext_vector_type(2)));
typedef float v8f __attribute__((ext_vector_type(8)));

constexpr int  BB   = 2;
constexpr int  NN   = 50000;
constexpr int  EE   = 800000;
constexpr int  INF  = 128;
constexpr int  HID  = 64;
constexpr long ROWS = (long)BB * NN;   // 100000
constexpr float EPSV = 1e-5f;

// ---------------- small utility kernels ----------------

__global__ void fill_val(float* __restrict__ p, long n, float v) {
  long i = (long)blockIdx.x * blockDim.x + threadIdx.x;
  if (i < n) p[i] = v;
}

// deg[c] += exp(ew[e]) for each edge (deg pre-filled with 1.0 for self loops)
__global__ void deg_acc(const float* __restrict__ ew, const int* __restrict__ ei,
                        float* __restrict__ deg) {
  long e = (long)blockIdx.x * blockDim.x + threadIdx.x;
  if (e < EE) atomicAdd(&deg[ei[EE + e]], expf(ew[e]));
}

__global__ void rsqrt_inplace(float* __restrict__ p, long n) {
  long i = (long)blockIdx.x * blockDim.x + threadIdx.x;
  if (i < n) p[i] = rsqrtf(p[i]);
}

// norm[e] = dinv[row] * exp(ew[e]) * dinv[col]
__global__ void norm_kernel(const float* __restrict__ ew, const int* __restrict__ ei,
                            const float* __restrict__ dinv, float* __restrict__ nrm) {
  long e = (long)blockIdx.x * blockDim.x + threadIdx.x;
  if (e < EE) nrm[e] = dinv[ei[e]] * expf(ew[e]) * dinv[ei[EE + e]];
}

// ---------------- fp32 WMMA GEMM:  C[ROWS x 64] = A[ROWS x K] * W[K x 64] ----------------
// Block = 128 threads = 4 waves; block owns 16 rows, wave w owns cols [16w,16w+16).
// A tile staged in LDS (stride K+1 to avoid bank conflicts); optional fused
// BatchNorm affine (scale/shift per input channel) applied at LDS-fill time.
template <int K, bool BN>
__global__ void gemm16x64_wmma(const float* __restrict__ A,
                               const float* __restrict__ W,
                               const float* __restrict__ scsh,  // [scale 0..63 | shift 64..127]
                               float* __restrict__ C) {
  __shared__ float As[16 * (K + 1)];
  const long rowBase = (long)blockIdx.x * 16;

  // cooperative, coalesced A-tile load (+ fused BN affine)
  for (int idx = threadIdx.x; idx < 16 * K; idx += 128) {
    int r = idx / K, kk = idx - r * K;
    float v = A[(rowBase + r) * K + kk];
    if (BN) v = v * scsh[kk] + scsh[64 + kk];
    As[r * (K + 1) + kk] = v;
  }
  __syncthreads();

  const int lane = threadIdx.x & 31;
  const int wave = threadIdx.x >> 5;     // column tile 0..3
  const int half = lane >> 4;            // 0: K%4 in {0,1}, 1: K%4 in {2,3}
  const int lm   = lane & 15;
  const int col  = wave * 16 + lm;

  v8f acc = {};
  for (int k = 0; k < K; k += 4) {
    const int ka = k + 2 * half;
    v2f a, b;
    // A 16x4 fp32 fragment: lanes0-15 K=k,k+1; lanes16-31 K=k+2,k+3 (M = lm)
    a.x = As[lm * (K + 1) + ka];
    a.y = As[lm * (K + 1) + ka + 1];
    // B 4x16 fp32 fragment (mirror layout, N = lm)
    b.x = W[ka * HID + col];
    b.y = W[(ka + 1) * HID + col];
    acc = __builtin_amdgcn_wmma_f32_16x16x4_f32(false, a, false, b,
                                                (short)0, acc, false, false);
  }

  // C 16x16 fp32: VGPR g holds M = g (lanes0-15) / M = g+8 (lanes16-31), N = lm
  const long rb = rowBase + 8 * half;
#pragma unroll
  for (int g = 0; g < 8; ++g)
    C[(rb + g) * HID + col] = acc[g];
}

// ---------------- message passing ----------------

// agg[b,n,c] = dinv[n]^2 * h[b,n,c]   (self-loop contribution, weight 1)
__global__ void self_loop_init(const float* __restrict__ h, const float* __restrict__ dinv,
                               float* __restrict__ agg) {
  long i = (long)blockIdx.x * blockDim.x + threadIdx.x;
  if (i >= ROWS * HID) return;
  long p = i >> 6;                 // b*N + n
  int n = (int)(p % NN);
  float d = dinv[n];
  agg[i] = d * d * h[i];
}

// agg[b,col,:] += norm[e] * h[b,row,:]  — 64 threads per (edge,batch) pair
__global__ void edge_scatter(const float* __restrict__ h, const int* __restrict__ ei,
                             const float* __restrict__ nrm, float* __restrict__ agg) {
  long i = (long)blockIdx.x * blockDim.x + threadIdx.x;
  if (i >= (long)EE * BB * HID) return;
  int  c = (int)(i & 63);
  long p = i >> 6;
  int  b = (int)(p & 1);
  int  e = (int)(p >> 1);
  int  r = ei[e], cl = ei[EE + e];
  float v = nrm[e] * h[((long)b * NN + r) * HID + c];
  atomicAdd(&agg[((long)b * NN + cl) * HID + c], v);
}

// h = relu(h + bias) in place; accumulate per-channel sum & sumsq into stats[128]
__global__ void bias_relu_stats(float* __restrict__ h, const float* __restrict__ bias,
                                float* __restrict__ stats) {
  const int c  = threadIdx.x & 63;
  const int rg = threadIdx.x >> 6;   // 4 row-groups per block
  const float bv = bias[c];
  float s = 0.f, q = 0.f;
  for (long r = (long)blockIdx.x * 4 + rg; r < ROWS; r += (long)gridDim.x * 4) {
    float v = h[r * HID + c] + bv;
    v = v > 0.f ? v : 0.f;
    h[r * HID + c] = v;
    s += v; q += v * v;
  }
  __shared__ float ssum[256], ssq[256];
  ssum[threadIdx.x] = s; ssq[threadIdx.x] = q;
  __syncthreads();
  if (rg == 0) {
    float ts = ssum[c] + ssum[c + 64] + ssum[c + 128] + ssum[c + 192];
    float tq = ssq[c]  + ssq[c + 64]  + ssq[c + 128]  + ssq[c + 192];
    atomicAdd(&stats[c], ts);
    atomicAdd(&stats[64 + c], tq);
  }
}

// scsh[c] = g*rsqrt(var+eps), scsh[64+c] = be - mean*scale   (64 threads, 1 block)
__global__ void bn_finalize(const float* __restrict__ stats, const float* __restrict__ g,
                            const float* __restrict__ be, float* __restrict__ scsh) {
  int c = threadIdx.x;
  float inv_n = 1.0f / (float)ROWS;
  float mean = stats[c] * inv_n;
  float var  = stats[64 + c] * inv_n - mean * mean;
  float sc   = g[c] * rsqrtf(var + EPSV);
  scsh[c] = sc;
  scsh[64 + c] = be[c] - mean * sc;
}

// out[r] = dot(bn2(h[r,:]), Wc) + bc  — BN2 fused; 64 threads per row, 4 rows/block
__global__ void classifier(const float* __restrict__ h, const float* __restrict__ scsh,
                           const float* __restrict__ Wc, const float* __restrict__ bc,
                           float* __restrict__ out) {
  __shared__ float red[256];
  const int c  = threadIdx.x & 63;
  const int rg = threadIdx.x >> 6;
  const long r = (long)blockIdx.x * 4 + rg;
  float v = 0.f;
  if (r < ROWS) v = (h[r * HID + c] * scsh[c] + scsh[64 + c]) * Wc[c];
  red[threadIdx.x] = v;
  __syncthreads();
  for (int off = 32; off > 0; off >>= 1) {
    if (c < off) red[threadIdx.x] += red[threadIdx.x + off];
    __syncthreads();
  }
  if (c == 0 && r < ROWS) out[r] = red[threadIdx.x] + bc[0];
}

// ---------------- orchestration ----------------

extern "C" void kernel_launch(void* const* d_in, const int* in_sizes, int n_in,
                              void* d_out, int out_size, void* d_ws, size_t ws_size,
                              hipStream_t stream) {
  const float* x   = (const float*)d_in[0];
  const float* ew  = (const float*)d_in[1];
  const float* W1  = (const float*)d_in[2];
  const float* b1  = (const float*)d_in[3];
  const float* W2  = (const float*)d_in[4];
  const float* b2  = (const float*)d_in[5];
  const float* g1  = (const float*)d_in[6];
  const float* be1 = (const float*)d_in[7];
  const float* g2  = (const float*)d_in[8];
  const float* be2 = (const float*)d_in[9];
  const float* Wc  = (const float*)d_in[10];
  const float* bc  = (const float*)d_in[11];
  const int*   ei  = (const int*)d_in[12];   // [2, E] (row, col) — per harness int ABI
  float* out = (float*)d_out;

  float* ws    = (float*)d_ws;
  float* dinv  = ws;                    // N        (holds deg, then 1/sqrt(deg))
  float* nrm   = dinv + NN;             // E
  float* bufA  = nrm + EE;              // ROWS*HID
  float* bufB  = bufA + ROWS * HID;     // ROWS*HID
  float* stats = bufB + ROWS * HID;     // 128 (sum | sumsq)
  float* scsh  = stats + 128;           // 128 (scale | shift)

  const int T = 256;
  auto nb = [](long n, int t) { return (int)((n + t - 1) / t); };

  // gcn_norm (with self-loops, fill_value = 1)
  fill_val<<<nb(NN, T), T, 0, stream>>>(dinv, NN, 1.0f);
  deg_acc<<<nb(EE, T), T, 0, stream>>>(ew, ei, dinv);
  rsqrt_inplace<<<nb(NN, T), T, 0, stream>>>(dinv, NN);
  norm_kernel<<<nb(EE, T), T, 0, stream>>>(ew, ei, dinv, nrm);

  // conv1: x @ W1 -> bufA; aggregate -> bufB; relu+stats; BN1 params -> scsh
  gemm16x64_wmma<INF, false><<<(int)(ROWS / 16), 128, 0, stream>>>(x, W1, nullptr, bufA);
  self_loop_init<<<nb(ROWS * HID, T), T, 0, stream>>>(bufA, dinv, bufB);
  edge_scatter<<<nb((long)EE * BB * HID, T), T, 0, stream>>>(bufA, ei, nrm, bufB);
  fill_val<<<1, 128, 0, stream>>>(stats, 128, 0.0f);
  bias_relu_stats<<<1024, 256, 0, stream>>>(bufB, b1, stats);
  bn_finalize<<<1, 64, 0, stream>>>(stats, g1, be1, scsh);

  // conv2: bn1(bufB) @ W2 -> bufA (BN fused into A load); aggregate -> bufB; relu+stats
  gemm16x64_wmma<HID, true><<<(int)(ROWS / 16), 128, 0, stream>>>(bufB, W2, scsh, bufA);
  self_loop_init<<<nb(ROWS * HID, T), T, 0, stream>>>(bufA, dinv, bufB);
  edge_scatter<<<nb((long)EE * BB * HID, T), T, 0, stream>>>(bufA, ei, nrm, bufB);
  fill_val<<<1, 128, 0, stream>>>(stats, 128, 0.0f);
  bias_relu_stats<<<1024, 256, 0, stream>>>(bufB, b2, stats);
  bn_finalize<<<1, 64, 0, stream>>>(stats, g2, be2, scsh);

  // classifier with BN2 fused
  classifier<<<(int)((ROWS + 3) / 4), 256, 0, stream>>>(bufB, scsh, Wc, bc, out);
}